// AdaptiveSubgraphLayer_19078244729196
// MI455X (gfx1250) — compile-verified
//
#include <hip/hip_runtime.h>
#include <math.h>
#include <float.h>

typedef __attribute__((ext_vector_type(2))) float v2f;
typedef __attribute__((ext_vector_type(8))) float v8f;

#define DIM 128

__device__ __forceinline__ int imin(int a, int b) { return a < b ? a : b; }

__device__ __forceinline__ float wred32(float v) {
#pragma unroll
  for (int m = 16; m > 0; m >>= 1) v += __shfl_xor(v, m, 32);
  return v;
}

// float atomic max/min via integer atomics (native global_atomic_{max,min}_{i32,u32})
__device__ __forceinline__ void atomicMaxF(float* a, float v) {
  if (v >= 0.0f) atomicMax((int*)a, __float_as_int(v));
  else           atomicMin((unsigned int*)a, (unsigned int)__float_as_int(v));
}
__device__ __forceinline__ void atomicMinF(float* a, float v) {
  if (v >= 0.0f) atomicMin((int*)a, __float_as_int(v));
  else           atomicMax((unsigned int*)a, (unsigned int)__float_as_int(v));
}

// ---------------------------------------------------------------------------
// K0: initialize accumulators.
// agg layout per node: [sum(128) | max(128) | min(128) | sumsq(128)]
// ---------------------------------------------------------------------------
__global__ void k_init(float* __restrict__ agg, float* __restrict__ deg,
                       float* __restrict__ hprev, int* __restrict__ idxb, int N) {
  long long Tagg = (long long)N * 512;
  long long Tdeg = Tagg + N;
  long long Thp  = Tdeg + (long long)N * DIM;
  long long T    = Thp + 16;
  long long stride = (long long)gridDim.x * blockDim.x;
  for (long long i = (long long)blockIdx.x * blockDim.x + threadIdx.x; i < T; i += stride) {
    if (i < Tagg) {
      int slab = ((int)(i >> 7)) & 3;
      float v = 0.0f;
      if (slab == 1) v = -FLT_MAX;
      else if (slab == 2) v = FLT_MAX;
      agg[i] = v;
    } else if (i < Tdeg) {
      deg[i - Tagg] = 0.0f;
    } else if (i < Thp) {
      hprev[i - Tdeg] = 0.0f;
    } else {
      idxb[i - Thp] = 0x7FFFFFFF;
    }
  }
}

// ---------------------------------------------------------------------------
// K1: h_prev[old_nodes_new_idx[p]] = hidden[p]   (float4 per lane)
// ---------------------------------------------------------------------------
__global__ void k_scatter_hprev(const float* __restrict__ hidden, const int* __restrict__ oni,
                                float* __restrict__ hprev, int nprev) {
  int t = blockIdx.x * blockDim.x + threadIdx.x;
  int p = t >> 5, q = t & 31;
  if (p >= nprev) return;
  int dst = oni[p];
  const float4* s = (const float4*)(hidden + (size_t)p * DIM);
  float4* d = (float4*)(hprev + (size_t)dst * DIM);
  d[q] = s[q];
}

// ---------------------------------------------------------------------------
// K2: per-edge hyperbolic message + atomic scatter of 4 statistics.
// One wave32 per edge, 4 dims per lane.
// ---------------------------------------------------------------------------
__global__ void k_edges(const float* __restrict__ hidden, const float* __restrict__ rela,
                        const int* __restrict__ edges, float* __restrict__ agg,
                        float* __restrict__ deg, int nedges) {
  int wid = (int)((blockIdx.x * (long long)blockDim.x + threadIdx.x) >> 5);
  int lane = threadIdx.x & 31;
  if (wid >= nedges) return;
  const int* e = edges + (size_t)wid * 6;
  int sub = e[4], rel = e[2], obj = e[5];

  float4 u = ((const float4*)(hidden + (size_t)sub * DIM))[lane];
  float4 w = ((const float4*)(rela   + (size_t)rel * DIM))[lane];

  // expmap0 + project for hs -> x (scale sx) and hr -> y (scale sy)
  float du = wred32(u.x * u.x + u.y * u.y + u.z * u.z + u.w * u.w);
  float dw = wred32(w.x * w.x + w.y * w.y + w.z * w.z + w.w * w.w);
  float uw = wred32(u.x * w.x + u.y * w.y + u.z * w.z + u.w * w.w);

  float nu = sqrtf(du);
  float n1 = fmaxf(nu, 1e-15f);
  float t1 = tanhf(fminf(n1, 15.0f));
  float s1 = t1 / n1;
  float gnx = s1 * nu;                                  // ||expmap0(u)||
  float sx = s1 * ((gnx > 0.996f) ? (0.996f / fmaxf(gnx, 1e-15f)) : 1.0f);

  float nw = sqrtf(dw);
  float n2_ = fmaxf(nw, 1e-15f);
  float t2 = tanhf(fminf(n2_, 15.0f));
  float s2 = t2 / n2_;
  float gny = s2 * nw;
  float sy = s2 * ((gny > 0.996f) ? (0.996f / fmaxf(gny, 1e-15f)) : 1.0f);

  // mobius_add(x, y), c = 1
  float x2 = sx * sx * du;
  float y2 = sy * sy * dw;
  float xy = sx * sy * uw;
  float ca = (1.0f + 2.0f * xy + y2) * sx;
  float cb = (1.0f - x2) * sy;
  float den = fmaxf(1.0f + 2.0f * xy + x2 * y2, 1e-15f);
  float rden = 1.0f / den;

  float m0 = (ca * u.x + cb * w.x) * rden;
  float m1 = (ca * u.y + cb * w.y) * rden;
  float m2 = (ca * u.z + cb * w.z) * rden;
  float m3 = (ca * u.w + cb * w.w) * rden;

  // project + logmap0
  float r2 = wred32(m0 * m0 + m1 * m1 + m2 * m2 + m3 * m3);
  float r = sqrtf(r2);
  float sp = (r > 0.996f) ? (0.996f / fmaxf(r, 1e-15f)) : 1.0f;
  float pn = fmaxf(r * sp, 1e-15f);
  float z = fminf(pn, 1.0f - 1e-5f);
  float a = atanhf(z);
  float sc = sp * a / pn;
  m0 *= sc; m1 *= sc; m2 *= sc; m3 *= sc;

  float* base = agg + (size_t)obj * 512;
  int d = lane * 4;
  // sum slab (+0)
  atomicAdd(base + d + 0, m0);
  atomicAdd(base + d + 1, m1);
  atomicAdd(base + d + 2, m2);
  atomicAdd(base + d + 3, m3);
  // sumsq slab (+384)
  atomicAdd(base + 384 + d + 0, m0 * m0);
  atomicAdd(base + 384 + d + 1, m1 * m1);
  atomicAdd(base + 384 + d + 2, m2 * m2);
  atomicAdd(base + 384 + d + 3, m3 * m3);
  // max slab (+128)
  atomicMaxF(base + 128 + d + 0, m0);
  atomicMaxF(base + 128 + d + 1, m1);
  atomicMaxF(base + 128 + d + 2, m2);
  atomicMaxF(base + 128 + d + 3, m3);
  // min slab (+256)
  atomicMinF(base + 256 + d + 0, m0);
  atomicMinF(base + 256 + d + 1, m1);
  atomicMinF(base + 256 + d + 2, m2);
  atomicMinF(base + 256 + d + 3, m3);
  if (lane == 0) atomicAdd(deg + obj, 1.0f);
}

// ---------------------------------------------------------------------------
// K3: finalize agg in place: [mean | max' | min' | std], compute amp/att.
// ---------------------------------------------------------------------------
__global__ void k_finalize(float* __restrict__ agg, const float* __restrict__ deg,
                           float* __restrict__ amp, float* __restrict__ att, int N) {
  int i = blockIdx.x * blockDim.x + threadIdx.x;
  if (i >= N * DIM) return;
  int n = i >> 7, d = i & 127;
  float* b = agg + (size_t)n * 512;
  float dg = deg[n];
  float d1 = fmaxf(dg, 1.0f);
  float s  = b[d];
  float mx = b[128 + d];
  float mn = b[256 + d];
  float s2 = b[384 + d];
  float mean = s / d1;
  float m2 = s2 / d1;
  float sd = sqrtf(fmaxf(m2 - mean * mean, 0.0f) + 1e-10f);
  bool has = dg > 0.0f;
  b[d]        = mean;
  b[128 + d]  = has ? mx : 0.0f;
  b[256 + d]  = has ? mn : 0.0f;
  b[384 + d]  = sd;
  if (d == 0) {
    float logd = log1pf(dg);       // PNA_DELTA = 1
    amp[n] = logd;
    att[n] = 1.0f / fmaxf(logd, 1e-5f);
  }
}

// ---------------------------------------------------------------------------
// K4: h_tilde = feat @ W_agg + b_agg using V_WMMA_F32_16X16X4_F32.
// feat = [agg | agg*amp | agg*att | h_prev]; amp/att folded into epilogue via
// 3 accumulators. Block: 256 threads = 8 waves; tile M=32 x N=128; each wave
// owns 16 cols and 2 row-tiles.
// ---------------------------------------------------------------------------
__global__ void __launch_bounds__(256) k_gemm(
    const float* __restrict__ agg, const float* __restrict__ hprev,
    const float* __restrict__ W, const float* __restrict__ bias,
    const float* __restrict__ amp, const float* __restrict__ att,
    float* __restrict__ out, int N) {
  int wave = threadIdx.x >> 5;
  int lane = threadIdx.x & 31;
  int j0 = wave * 16;
  int m0 = blockIdx.x * 32;
  int mrow = lane & 15;
  int khalf = (lane >> 4) << 1;     // 0 or 2
  int ncol = j0 + (lane & 15);

  v8f aA0 = {0,0,0,0,0,0,0,0}, aB0 = {0,0,0,0,0,0,0,0}, aC0 = {0,0,0,0,0,0,0,0};
  v8f aA1 = {0,0,0,0,0,0,0,0}, aB1 = {0,0,0,0,0,0,0,0}, aC1 = {0,0,0,0,0,0,0,0};

  const float* row0 = agg + (size_t)imin(m0 + mrow,      N - 1) * 512;
  const float* row1 = agg + (size_t)imin(m0 + 16 + mrow, N - 1) * 512;

  for (int k = 0; k < 512; k += 4) {
    int kk = k + khalf;
    v2f a0 = { row0[kk], row0[kk + 1] };
    v2f a1 = { row1[kk], row1[kk + 1] };
    v2f b0 = { W[(size_t)kk * 128 + ncol],          W[(size_t)(kk + 1) * 128 + ncol] };
    v2f b1 = { W[(size_t)(512 + kk) * 128 + ncol],  W[(size_t)(513 + kk) * 128 + ncol] };
    v2f b2 = { W[(size_t)(1024 + kk) * 128 + ncol], W[(size_t)(1025 + kk) * 128 + ncol] };
    aA0 = __builtin_amdgcn_wmma_f32_16x16x4_f32(false, a0, false, b0, (short)0, aA0, false, false);
    aB0 = __builtin_amdgcn_wmma_f32_16x16x4_f32(false, a0, false, b1, (short)0, aB0, false, false);
    aC0 = __builtin_amdgcn_wmma_f32_16x16x4_f32(false, a0, false, b2, (short)0, aC0, false, false);
    aA1 = __builtin_amdgcn_wmma_f32_16x16x4_f32(false, a1, false, b0, (short)0, aA1, false, false);
    aB1 = __builtin_amdgcn_wmma_f32_16x16x4_f32(false, a1, false, b1, (short)0, aB1, false, false);
    aC1 = __builtin_amdgcn_wmma_f32_16x16x4_f32(false, a1, false, b2, (short)0, aC1, false, false);
  }

  const float* hp0 = hprev + (size_t)imin(m0 + mrow,      N - 1) * DIM;
  const float* hp1 = hprev + (size_t)imin(m0 + 16 + mrow, N - 1) * DIM;
  for (int k = 0; k < DIM; k += 4) {
    int kk = k + khalf;
    v2f a0 = { hp0[kk], hp0[kk + 1] };
    v2f a1 = { hp1[kk], hp1[kk + 1] };
    v2f b3 = { W[(size_t)(1536 + kk) * 128 + ncol], W[(size_t)(1537 + kk) * 128 + ncol] };
    aA0 = __builtin_amdgcn_wmma_f32_16x16x4_f32(false, a0, false, b3, (short)0, aA0, false, false);
    aA1 = __builtin_amdgcn_wmma_f32_16x16x4_f32(false, a1, false, b3, (short)0, aA1, false, false);
  }

  float bn = bias[ncol];
  int rowoff = (lane >> 4) << 3;    // 0 or 8
#pragma unroll
  for (int r = 0; r < 8; ++r) {
    int mA = m0 + r + rowoff;
    int mB = m0 + 16 + r + rowoff;
    if (mA < N) {
      float v = aA0[r] + amp[mA] * aB0[r] + att[mA] * aC0[r] + bn;
      out[(size_t)mA * 128 + ncol] = v;
    }
    if (mB < N) {
      float v = aA1[r] + amp[mB] * aB1[r] + att[mB] * aC1[r] + bn;
      out[(size_t)mB * 128 + ncol] = v;
    }
  }
}

// ---------------------------------------------------------------------------
// K5a: per batch b: idx[b] = min{ n : nodes[n,0]==b && nodes[n,1]==q_sub[b] }
// ---------------------------------------------------------------------------
__global__ void k_match(const int* __restrict__ nodes, const int* __restrict__ qsub,
                        int* __restrict__ idxb, int N, int B) {
  int n = blockIdx.x * blockDim.x + threadIdx.x;
  if (n >= N) return;
  int b = nodes[(size_t)n * 2];
  if (b >= 0 && b < B && nodes[(size_t)n * 2 + 1] == qsub[b]) atomicMin(&idxb[b], n);
}

// ---------------------------------------------------------------------------
// K5b: su[b] = dot(h_tilde[idx[b]], W_score[0:128])   (one wave per batch)
// ---------------------------------------------------------------------------
__global__ void k_user(const float* __restrict__ htilde, const float* __restrict__ Ws,
                       const int* __restrict__ idxb, float* __restrict__ su, int B, int N) {
  int wave = threadIdx.x >> 5, lane = threadIdx.x & 31;
  int nw = blockDim.x >> 5;
  for (int b = wave; b < B; b += nw) {
    int i = idxb[b];
    if (i < 0 || i >= N) i = 0;   // argmax of all-false -> 0
    float4 h  = ((const float4*)(htilde + (size_t)i * DIM))[lane];
    float4 ws = ((const float4*)Ws)[lane];
    float d = wred32(h.x * ws.x + h.y * ws.y + h.z * ws.z + h.w * ws.w);
    if (lane == 0) su[b] = d;
  }
}

// ---------------------------------------------------------------------------
// K6: alpha = sigmoid(su[nb] + dot(h_tilde[n], W_score[128:256]) + b_score);
//     out[n] = alpha * h_tilde[n]   (in place on d_out)
// ---------------------------------------------------------------------------
__global__ void k_score(float* __restrict__ htilde, const float* __restrict__ Ws,
                        const float* __restrict__ bs, const int* __restrict__ nodes,
                        const float* __restrict__ su, int N) {
  int wid = (int)((blockIdx.x * (long long)blockDim.x + threadIdx.x) >> 5);
  int lane = threadIdx.x & 31;
  if (wid >= N) return;
  float* row = htilde + (size_t)wid * DIM;
  float4 h  = ((const float4*)row)[lane];
  float4 ws = ((const float4*)(Ws + DIM))[lane];
  float d = wred32(h.x * ws.x + h.y * ws.y + h.z * ws.z + h.w * ws.w);
  int b = nodes[(size_t)wid * 2];
  float s = su[b] + d + bs[0];                 // TAU = 1
  float alpha = 1.0f / (1.0f + expf(-s));
  float4 o = { h.x * alpha, h.y * alpha, h.z * alpha, h.w * alpha };
  ((float4*)row)[lane] = o;
}

// ---------------------------------------------------------------------------
extern "C" void kernel_launch(void* const* d_in, const int* in_sizes, int n_in,
                              void* d_out, int out_size, void* d_ws, size_t ws_size,
                              hipStream_t stream) {
  const float* hidden = (const float*)d_in[0];
  const float* rela   = (const float*)d_in[1];
  const float* W_agg  = (const float*)d_in[2];
  const float* b_agg  = (const float*)d_in[3];
  const float* W_sc   = (const float*)d_in[4];
  const float* b_sc   = (const float*)d_in[5];
  const int*   edges  = (const int*)d_in[6];
  const int*   nodes  = (const int*)d_in[7];
  const int*   qsub   = (const int*)d_in[8];
  const int*   oni    = (const int*)d_in[9];

  const int nprev  = in_sizes[0] / DIM;
  const int nedges = in_sizes[6] / 6;
  const int N      = in_sizes[7] / 2;
  const int B      = in_sizes[8];

  float* agg   = (float*)d_ws;                 // N*512
  float* hprev = agg + (size_t)N * 512;        // N*128
  float* deg   = hprev + (size_t)N * DIM;      // N
  float* amp   = deg + N;                      // N
  float* att   = amp + N;                      // N
  float* su    = att + N;                      // B
  int*   idxb  = (int*)(su + B);               // B (<=16)
  float* htilde = (float*)d_out;               // N*128 lives in d_out

  k_init<<<4096, 256, 0, stream>>>(agg, deg, hprev, idxb, N);
  k_scatter_hprev<<<(nprev * 32 + 255) / 256, 256, 0, stream>>>(hidden, oni, hprev, nprev);
  k_edges<<<(nedges + 7) / 8, 256, 0, stream>>>(hidden, rela, edges, agg, deg, nedges);
  k_match<<<(N + 255) / 256, 256, 0, stream>>>(nodes, qsub, idxb, N, B);
  k_finalize<<<(N * DIM + 255) / 256, 256, 0, stream>>>(agg, deg, amp, att, N);
  k_gemm<<<(N + 31) / 32, 256, 0, stream>>>(agg, hprev, W_agg, b_agg, amp, att, htilde, N);
  k_user<<<1, 256, 0, stream>>>(htilde, W_sc, idxb, su, B, N);
  k_score<<<(N + 7) / 8, 256, 0, stream>>>(htilde, W_sc, b_sc, nodes, su, N);
}